// InteractBlock_76510547411400
// MI455X (gfx1250) — compile-verified
//
#include <hip/hip_runtime.h>

// MI455X / gfx1250 (CDNA5, wave32). One wave per edge:
//   A[p,q] = sum_r cg[p,q,r] * Y[r]           (54 FMAs, cg preloaded in WMMA-A layout)
//   mixed  = A(9x9) @ x_j(9x32)               (6x v_wmma_f32_16x16x4_f32, K padded to 12)
//   msg    = msg_a + gate * mixed             (register epilogue from WMMA D layout)
//   out[dst] += msg                           (native f32 L2 atomics; out/x are L2-resident)

typedef __attribute__((ext_vector_type(2))) float v2f;
typedef __attribute__((ext_vector_type(8))) float v8f;

#define SDIM 9
#define FDIM 32

__device__ __forceinline__ float envelope_f(float d) {
    float u  = fminf(fmaxf(d, 0.0f), 1.0f);
    float u2 = u * u;
    float u4 = u2 * u2;
    float u5 = u4 * u;
    // 1 - 21 u^5 + 35 u^6 - 15 u^7
    return 1.0f - 21.0f * u5 + 35.0f * u5 * u - 15.0f * u5 * u2;
}

__global__ void __launch_bounds__(256) init_out_kernel(const float4* __restrict__ x,
                                                       float4* __restrict__ out,
                                                       int n4) {
    int i = blockIdx.x * blockDim.x + threadIdx.x;
    int stride = gridDim.x * blockDim.x;
    for (; i < n4; i += stride) out[i] = x[i];
}

__global__ void __launch_bounds__(256) edge_msg_kernel(
    const float* __restrict__ x,        // (N, 9, 32)
    const int*   __restrict__ ei,       // (2, E)
    const float* __restrict__ ea,       // (E, 9, 2): [...,0]=Y, [...,1]=dist
    const float* __restrict__ Wsca,     // (3, 32)
    const float* __restrict__ Wsph,     // (3, 32)
    const float* __restrict__ Wmix,     // (3, 32)
    const float* __restrict__ cg,       // (9, 9, 9)
    float* __restrict__ out,            // (N, 9, 32), pre-seeded with x
    int E) {

    const int lane = threadIdx.x & 31;
    const int wid  = threadIdx.x >> 5;
    const int half = lane >> 4;     // 0: lanes 0-15, 1: lanes 16-31
    const int col  = lane & 15;     // WMMA column (f) / A-matrix row (p)

    // ---- Preload cg permuted into WMMA-A striping (loaded once, reused for all edges).
    // A 16x4 f32 layout: lane m holds A[m, 2*half + j] in component j (q = 4*kc + 2*half + j).
    float cgreg[3][2][SDIM];
#pragma unroll
    for (int kc = 0; kc < 3; ++kc) {
#pragma unroll
        for (int j = 0; j < 2; ++j) {
            const int q  = 4 * kc + 2 * half + j;
            const bool ok = (col < SDIM) && (q < SDIM);
            const int pc = (col < SDIM) ? col : 0;    // clamp -> always in-bounds load
            const int qc = (q   < SDIM) ? q   : 0;
#pragma unroll
            for (int r = 0; r < SDIM; ++r) {
                float v = cg[(pc * SDIM + qc) * SDIM + r];
                cgreg[kc][j][r] = ok ? v : 0.0f;
            }
        }
    }

    // ---- Preload per-(l, f-tile) weights.
    float wsca[3][2], wsph[3][2], wmix[3][2];
#pragma unroll
    for (int l = 0; l < 3; ++l) {
#pragma unroll
        for (int t = 0; t < 2; ++t) {
            const int f = col + 16 * t;
            wsca[l][t] = Wsca[l * FDIM + f];
            wsph[l][t] = Wsph[l * FDIM + f];
            wmix[l][t] = Wmix[l * FDIM + f];
        }
    }

    const int wavesPerBlk = blockDim.x >> 5;
    const int waveStride  = gridDim.x * wavesPerBlk;

    for (int e0 = blockIdx.x * wavesPerBlk + wid; e0 < E; e0 += waveStride) {
        const int e   = __builtin_amdgcn_readfirstlane(e0);    // wave-uniform -> SMEM path
        const int src = ei[e];
        const int dst = ei[E + e];

        const float* eap = ea + (size_t)e * (SDIM * 2);
        float Y[16];                       // padded so Y[v+8] is always a legal index
#pragma unroll
        for (int r = 0; r < SDIM; ++r) Y[r] = eap[2 * r];
#pragma unroll
        for (int r = SDIM; r < 16; ++r) Y[r] = 0.0f;
        const float envd = envelope_f(eap[1]);
        const float envy = envelope_f(Y[0]);

        // ---- A[p,q] = sum_r cg[p,q,r] * Y[r], built directly in WMMA-A registers.
        v2f A[3];
#pragma unroll
        for (int kc = 0; kc < 3; ++kc) {
            float a0 = 0.0f, a1 = 0.0f;
#pragma unroll
            for (int r = 0; r < SDIM; ++r) {
                a0 = __builtin_fmaf(cgreg[kc][0][r], Y[r], a0);
                a1 = __builtin_fmaf(cgreg[kc][1][r], Y[r], a1);
            }
            A[kc].x = a0;
            A[kc].y = a1;
        }

        // ---- B = x_j in WMMA-B striping: lane col holds column f, rows q = 4*kc+2*half+j.
        const float* xj = x + (size_t)src * (SDIM * FDIM);
        v2f B[3][2];
#pragma unroll
        for (int kc = 0; kc < 3; ++kc) {
            const int q0  = 4 * kc + 2 * half;
            const int q1  = q0 + 1;
            const int q0c = (q0 < SDIM) ? q0 : 0;   // clamp: keep loads in-bounds
            const int q1c = (q1 < SDIM) ? q1 : 0;
#pragma unroll
            for (int t = 0; t < 2; ++t) {
                const int f = col + 16 * t;
                float b0 = xj[q0c * FDIM + f];
                float b1 = xj[q1c * FDIM + f];
                B[kc][t].x = (q0 < SDIM) ? b0 : 0.0f;
                B[kc][t].y = (q1 < SDIM) ? b1 : 0.0f;
            }
        }

        // ---- mixed = A @ x_j via exact-f32 WMMA (K=9 padded to 12 -> 3 k-steps, 2 f-tiles).
        v8f acc0 = {};
        v8f acc1 = {};
#pragma unroll
        for (int kc = 0; kc < 3; ++kc) {
            acc0 = __builtin_amdgcn_wmma_f32_16x16x4_f32(
                false, A[kc], false, B[kc][0], (short)0, acc0, false, false);
            acc1 = __builtin_amdgcn_wmma_f32_16x16x4_f32(
                false, A[kc], false, B[kc][1], (short)0, acc1, false, false);
        }

        // ---- Epilogue from D layout (VGPR v: rows p=v / p=v+8), scatter-add to out[dst].
        float* orow = out + (size_t)dst * (SDIM * FDIM);
#pragma unroll
        for (int v = 0; v < 8; ++v) {
            if (half == 0 || v == 0) {          // valid rows: p = 0..8 only
                const int p  = half ? (v + 8) : v;      // half==1 reachable only for v==0 (p=8)
                const int lv = (v == 0) ? 0 : ((v < 4) ? 1 : 2);   // l(p) for half==0; p=8 -> l=2
                const float Yp = half ? Y[v + 8] : Y[v];
#pragma unroll
                for (int t = 0; t < 2; ++t) {
                    const int f  = col + 16 * t;
                    const float ws = half ? wsca[2][t] : wsca[lv][t];
                    const float wh = half ? wsph[2][t] : wsph[lv][t];
                    const float wm = half ? wmix[2][t] : wmix[lv][t];
                    const float xv  = xj[p * FDIM + f];
                    const float mix = (t == 0) ? acc0[v] : acc1[v];
                    const float msg = xv * envd * ws * Yp * wh + envy * wm * mix;
                    __hip_atomic_fetch_add(orow + p * FDIM + f, msg,
                                           __ATOMIC_RELAXED, __HIP_MEMORY_SCOPE_AGENT);
                }
            }
        }
    }
}

extern "C" void kernel_launch(void* const* d_in, const int* in_sizes, int n_in,
                              void* d_out, int out_size, void* d_ws, size_t ws_size,
                              hipStream_t stream) {
    const float* x    = (const float*)d_in[0];
    const int*   ei   = (const int*)  d_in[1];
    const float* ea   = (const float*)d_in[2];
    const float* Wsca = (const float*)d_in[3];
    const float* Wsph = (const float*)d_in[4];
    const float* Wmix = (const float*)d_in[5];
    const float* cg   = (const float*)d_in[6];
    float* out = (float*)d_out;

    const int E = in_sizes[1] / 2;     // edge_index is (2, E)
    const int total = out_size;        // N * 9 * 32 (divisible by 4)

    // out = x (deterministic re-init every launch; edge kernel then accumulates).
    const int n4 = total / 4;
    int cblocks = (n4 + 255) / 256;
    init_out_kernel<<<cblocks, 256, 0, stream>>>((const float4*)x, (float4*)out, n4);

    // 8192 blocks x 8 waves = 65536 waves; grid-stride over 524288 edges (8 edges/wave).
    edge_msg_kernel<<<8192, 256, 0, stream>>>(x, ei, ea, Wsca, Wsph, Wmix, cg, out, E);
}